// ReLUcov_16982300688600
// MI455X (gfx1250) — compile-verified
//
#include <hip/hip_runtime.h>
#include <hip/hip_bf16.h>
#include <math.h>

// Shapes fixed by the reference: bs=4, n=256, c=96, d=2
#define BS 4
#define NN 256
#define CC 96

typedef __attribute__((ext_vector_type(2))) float v2f;
typedef __attribute__((ext_vector_type(8))) float v8f;

__device__ __forceinline__ float Phi_f(float x) {
    // 0.5*(1+erf(x/sqrt(2)))
    return 0.5f * (1.0f + erff(x * 0.70710678118654752440f));
}
__device__ __forceinline__ float phi_f(float x) {
    // (1/sqrt(2*pi)) * exp(-x^2/2) ; v_exp_f32 hardware path
    return 0.39894228040143267794f * __expf(-0.5f * x * x);
}

// ---------------------------------------------------------------------------
// Kernel A: Eh[b,i,k], dvals[b,i], xin passthrough.
// One block per (b,i), 96 threads (one per channel).
// ---------------------------------------------------------------------------
__global__ void eh_dvals_kernel(const float* __restrict__ xin,
                                const float* __restrict__ meanin,
                                const float* __restrict__ kin,
                                float* __restrict__ out,     // full d_out
                                float* __restrict__ dvals)   // ws: bs*n
{
    const int bi  = blockIdx.x;          // b*NN + i
    const int b   = bi / NN;
    const int i   = bi - b * NN;
    const int tid = threadIdx.x;         // 0..95

    __shared__ float red[CC];

    const float kd = kin[(size_t)b * NN * NN + (size_t)i * NN + i] + 1e-6f;
    const float sd = sqrtf(fmaxf(kd, 1e-8f));

    const float m  = meanin[(size_t)bi * CC + tid];
    const float z  = m / sd;
    const float Ph = Phi_f(z);
    const float ph = phi_f(z);
    const float eh = m * Ph + sd * ph;

    out[BS * NN * 2 + (size_t)bi * CC + tid] = eh;              // Eh region
    red[tid] = (m * m + sd * sd) * Ph + m * sd * ph - eh * eh;  // dval contrib
    __syncthreads();
    if (tid == 0) {
        float s = 0.0f;
        #pragma unroll 8
        for (int k = 0; k < CC; ++k) s += red[k];
        dvals[bi] = fmaxf(s * (1.0f / CC), 1e-6f);
    }
    // xin passthrough: first bs*n*2 = 2048 floats of d_out
    const int gid = bi * CC + tid;
    if (gid < BS * NN * 2) out[gid] = xin[gid];
}

// ---------------------------------------------------------------------------
// Kernel B: G[b,i,j] = (1/c) * sum_k Eh[b,i,k]*Eh[b,j,k] via f32 WMMA.
// One wave32 per 16x16 tile; 4 batches * 16*16 tiles = 1024 waves.
// A is 16x4 f32 (v2f/lane), B = Eh^T loads identically from Eh rows.
// ---------------------------------------------------------------------------
__global__ void gram_wmma_kernel(const float* __restrict__ eh,  // (bs,n,c)
                                 float* __restrict__ G)         // (bs,n,n)
{
    const int wave = (blockIdx.x * blockDim.x + threadIdx.x) >> 5;
    const int lane = threadIdx.x & 31;
    const int b    = wave >> 8;          // 0..3
    const int t    = wave & 255;         // tile within batch
    const int i0   = (t >> 4) << 4;
    const int j0   = (t & 15) << 4;
    const int half = lane >> 4;          // 0: K{0,1} / M{0..7}; 1: K{2,3} / M{8..15}
    const int l    = lane & 15;

    const float* arow = eh + (size_t)(b * NN + i0 + l) * CC;
    const float* brow = eh + (size_t)(b * NN + j0 + l) * CC;

    v8f acc = {0.f, 0.f, 0.f, 0.f, 0.f, 0.f, 0.f, 0.f};
    for (int k0 = 0; k0 < CC; k0 += 4) {
        // A 16x4 layout: VGPR0 = K0(lanes0-15)/K2(lanes16-31), VGPR1 = K1/K3
        v2f a  = *(const v2f*)(arow + k0 + 2 * half);
        v2f bm = *(const v2f*)(brow + k0 + 2 * half);
        acc = __builtin_amdgcn_wmma_f32_16x16x4_f32(
            /*neg_a=*/false, a, /*neg_b=*/false, bm,
            /*c_mod=*/(short)0, acc, /*reuse_a=*/false, /*reuse_b=*/false);
    }

    // C/D layout: VGPR v -> M = v + 8*half, N = l
    float* gt = G + (size_t)b * NN * NN;
    const float scale = 1.0f / CC;
    #pragma unroll
    for (int v = 0; v < 8; ++v) {
        gt[(size_t)(i0 + v + 8 * half) * NN + (j0 + l)] = acc[v] * scale;
    }
}

// ---------------------------------------------------------------------------
// Kernel C: the transcendental-heavy covar. One block per (b,i), thread = j.
// ---------------------------------------------------------------------------
__global__ void covar_kernel(const float* __restrict__ meanin,
                             const float* __restrict__ kin,
                             const float* __restrict__ dvals,
                             const float* __restrict__ G,
                             float* __restrict__ out3)   // (bs,n,n) region
{
    const int bi = blockIdx.x;           // b*NN + i
    const int b  = bi / NN;
    const int i  = bi - b * NN;
    const int j  = threadIdx.x;          // 0..255

    __shared__ __align__(16) float smi[CC];
    if (j < CC) smi[j] = meanin[(size_t)bi * CC + j];
    __syncthreads();

    const float* kb = kin + (size_t)b * NN * NN;
    const float eps_ij = (i == j) ? 1e-6f : 0.0f;
    const float m00 = kb[(size_t)i * NN + i] + 1e-6f;
    const float m11 = kb[(size_t)j * NN + j] + 1e-6f;
    const float m01 = kb[(size_t)i * NN + j] + eps_ij;
    const float m10 = kb[(size_t)j * NN + i] + eps_ij;

    const float det = m00 * m11 - m01 * m10;
    const float tr  = m00 + m11;
    const float s   = sqrtf(fmaxf(det, 1e-8f));
    const float t   = fmaxf(sqrtf(tr + 2.0f * s), 1e-8f);
    const float rt  = 1.0f / t;
    const float s00 = (m00 + s) * rt;
    const float s01 = m01 * rt;
    const float s10 = m10 * rt;
    const float s11 = (m11 + s) * rt;
    const float dS  = fmaxf(s00 * s11 - s01 * s10, 1e-8f);
    const float rdS = 1.0f / dS;
    const float i00 =  s11 * rdS;
    const float i01 = -s01 * rdS;
    const float i10 = -s10 * rdS;
    const float i11 =  s00 * rdS;

    const float* mjrow = meanin + (size_t)(b * NN + j) * CC;

    float acc = 0.0f;
    for (int k = 0; k < CC; k += 4) {
        const float4 mjv = *(const float4*)(mjrow + k);
        const float4 miv = *(const float4*)(&smi[k]);
        const float mis[4] = {miv.x, miv.y, miv.z, miv.w};
        const float mjs[4] = {mjv.x, mjv.y, mjv.z, mjv.w};
        #pragma unroll
        for (int u = 0; u < 4; ++u) {
            const float mi = mis[u];
            const float mj = mjs[u];
            const float u1 = i00 * mi + i01 * mj;
            const float u2 = i10 * mi + i11 * mj;
            const float Phi1 = Phi_f(u1);
            const float Phi2 = Phi_f(u2);
            const float p1 = phi_f(u1);
            const float p2 = phi_f(u2);
            float e = (mi * mj + m01) * (Phi1 * Phi2);           // EhhT_Psi
            e += (mj * s00 + mi * s10) * (p1 * Phi2);            // EhhT_psi1
            e += (mj * s01 + mi * s11) * (Phi1 * p2);            // EhhT_psi2
            e += s00 * s01 * (-u1 * p1) * Phi2                   // SHPhiS11
               + 2.0f * s01 * s01 * p1 * p2                      // 2*SHPhiS12
               + s01 * s11 * Phi1 * (-u2 * p2);                  // SHPhiS22
            acc += e;
        }
    }

    const float covar = acc * (1.0f / CC) - G[(size_t)bi * NN + j];
    out3[(size_t)bi * NN + j] = (i == j) ? (dvals[bi] + 1e-6f) : covar;
}

// ---------------------------------------------------------------------------
extern "C" void kernel_launch(void* const* d_in, const int* in_sizes, int n_in,
                              void* d_out, int out_size, void* d_ws, size_t ws_size,
                              hipStream_t stream) {
    const float* xin    = (const float*)d_in[0];   // (4,256,2)
    const float* meanin = (const float*)d_in[1];   // (4,256,96)
    const float* kin    = (const float*)d_in[2];   // (4,256,256)

    float* out   = (float*)d_out;
    float* dvals = (float*)d_ws;                   // bs*n       = 1024 floats
    float* G     = dvals + BS * NN;                // bs*n*n     = 262144 floats
    float* eh    = out + BS * NN * 2;              // Eh region of d_out
    float* out3  = out + BS * NN * 2 + BS * NN * CC; // covar region

    // A: Eh, dvals, xin passthrough
    eh_dvals_kernel<<<BS * NN, CC, 0, stream>>>(xin, meanin, kin, out, dvals);

    // B: Gram matrix via v_wmma_f32_16x16x4_f32 (1024 waves, 8 waves/block)
    gram_wmma_kernel<<<(BS * 256) / 8, 256, 0, stream>>>(eh, G);

    // C: transcendental-heavy covariance
    covar_kernel<<<BS * NN, NN, 0, stream>>>(meanin, kin, dvals, G, out3);
}